// MeanAggregator_79035988181013
// MI455X (gfx1250) — compile-verified
//
#include <hip/hip_runtime.h>

// MeanAggregator on MI455X (gfx1250, wave32).
// One wave per batch row: lane k owns neighbor slot k (K == warpSize == 32).
// Dedup via cross-lane shuffles; weighted mean via V_WMMA_F32_16X16X4_F32
// (the only full-precision f32 WMMA). A = gathered embed tile (16 d x 4 k),
// B = dedup weights broadcast along N, accumulated in f32 over 8 K-chunks.
// All gather base addresses + weights are hoisted out of the d-tile loop so
// the steady state is global_load_b32 (imm offset) + v_wmma only.

typedef __attribute__((ext_vector_type(2))) float v2f;
typedef __attribute__((ext_vector_type(8))) float v8f;

#define B_ 2048
#define K_ 32
#define N_ 50000
#define D_ 128
#define NCHUNK (K_ / 4)   // 8 K-chunks of 4 neighbor slots
#define NTILE  (D_ / 16)  // 8 d-tiles of 16 channels

__global__ __launch_bounds__(256) void mean_agg_wmma_f32(
    const float* __restrict__ embed,   // [N, D]
    const int*   __restrict__ nidx,    // [B, K]
    float*       __restrict__ out) {   // [B, D]
  const int lane = threadIdx.x & 31;
  const int wave = threadIdx.x >> 5;
  const int row  = blockIdx.x * 8 + wave;   // 256 threads = 8 waves = 8 rows

  // ---- load this row's 32 neighbor ids, one per lane (wave32 == K) ----
  const int myidx = nidx[row * K_ + lane];

  // Warm the cache hierarchy: each lane prefetches its neighbor row (512 B)
  // while the dedup shuffle scan below runs. gfx1250 global_prefetch_b8.
  {
    const float* r = embed + (size_t)myidx * D_;
    __builtin_prefetch(r,       0, 1);
    __builtin_prefetch(r +  32, 0, 1);
    __builtin_prefetch(r +  64, 0, 1);
    __builtin_prefetch(r +  96, 0, 1);
  }

  // ---- set-semantics dedup: lane is "first occurrence" iff no lower lane
  //      holds the same id. O(32) shuffle scan, no LDS. ----
  int dup = 0;
#pragma unroll
  for (int j = 0; j < K_; ++j) {
    int v = __shfl(myidx, j, 32);
    if (j < lane && v == myidx) dup = 1;
  }
  const int first = dup ? 0 : 1;

  // unique-neighbor count via butterfly reduction (all lanes get it)
  int cnt = first;
#pragma unroll
  for (int off = 16; off > 0; off >>= 1) cnt += __shfl_xor(cnt, off, 32);

  // dedup weight: 1/c for first occurrence, 0 for duplicates
  const float w = first ? (1.0f / (float)cnt) : 0.0f;

  // A-matrix 16x4 f32 layout (ISA 7.12.2): lanes 0-15 -> M=lane, V0=K0,V1=K1;
  // lanes 16-31 -> M=lane-16, V0=K2,V1=K3. B mirrors: lane halves hold
  // {K0,K1} / {K2,K3}, weights broadcast along N.
  const int half = (lane < 16) ? 0 : 2;   // K offset for this lane half
  const int m16  = lane & 15;             // M (d-channel within tile)

  // ---- hoist loop-invariant gather bases + B-operand weights ----
  const float* ap0[NCHUNK];
  const float* ap1[NCHUNK];
  float        b0v[NCHUNK];
  float        b1v[NCHUNK];
#pragma unroll
  for (int c = 0; c < NCHUNK; ++c) {
    const int s  = 4 * c + half;
    const int g0 = __shfl(myidx, s,     32);
    const int g1 = __shfl(myidx, s + 1, 32);
    ap0[c] = embed + (size_t)g0 * D_ + m16;
    ap1[c] = embed + (size_t)g1 * D_ + m16;
    b0v[c] = __shfl(w, s,     32);
    b1v[c] = __shfl(w, s + 1, 32);
  }

  // ---- 8 d-tiles; per tile: 16 gather loads (imm offsets) + 8 WMMAs ----
#pragma unroll
  for (int t = 0; t < NTILE; ++t) {
    float a0[NCHUNK], a1[NCHUNK];
#pragma unroll
    for (int c = 0; c < NCHUNK; ++c) {   // batch all 16 loads first
      a0[c] = ap0[c][16 * t];            // 64B-coalesced per operand row
      a1[c] = ap1[c][16 * t];
    }
    v8f acc = {0.f, 0.f, 0.f, 0.f, 0.f, 0.f, 0.f, 0.f};
#pragma unroll
    for (int c = 0; c < NCHUNK; ++c) {
      v2f Am = {a0[c], a1[c]};
      v2f Bm = {b0v[c], b1v[c]};
      // D[m][n] = sum_k embed[g_k][16t+m] * w_k  (identical across n)
      acc = __builtin_amdgcn_wmma_f32_16x16x4_f32(
          /*neg_a=*/false, Am, /*neg_b=*/false, Bm,
          /*c_mod=*/(short)0, acc, /*reuse_a=*/false, /*reuse_b=*/false);
    }
    // C/D layout: VGPR g = D[M=g][N=lane] (lanes 0-15), D[M=g+8][N=lane-16]
    // (lanes 16-31). Column N=0 lives in lanes 0 and 16: lane0 acc[g] =
    // out[16t+g], lane16 acc[g] = out[16t+8+g].
    if ((lane & 15) == 0) {
      float* p = out + (size_t)row * D_ + 16 * t + ((lane >> 1) & 8);
      ((float4*)p)[0] = make_float4(acc[0], acc[1], acc[2], acc[3]);
      ((float4*)p)[1] = make_float4(acc[4], acc[5], acc[6], acc[7]);
    }
  }
}

extern "C" void kernel_launch(void* const* d_in, const int* in_sizes, int n_in,
                              void* d_out, int out_size, void* d_ws, size_t ws_size,
                              hipStream_t stream) {
  const float* embed = (const float*)d_in[0];   // [N*D] f32
  const int*   nidx  = (const int*)d_in[1];     // [B*K] i32
  float*       out   = (float*)d_out;           // [B*D] f32

  dim3 block(256);              // 8 wave32 waves -> 8 batch rows per block
  dim3 grid(B_ / 8);            // 2048 rows -> 256 blocks
  hipLaunchKernelGGL(mean_agg_wmma_f32, grid, block, 0, stream,
                     embed, nidx, out);
}